// HeterogeneousOmicsGNN_36919538876536
// MI455X (gfx1250) — compile-verified
//
#include <hip/hip_runtime.h>
#include <math.h>

typedef __attribute__((ext_vector_type(2))) float v2f;
typedef __attribute__((ext_vector_type(4))) float v4f;
typedef __attribute__((ext_vector_type(8))) float v8f;

#define NNODES 50000
#define NEDGES 500000
#define EMB 128
#define HIDN 256

// ---------------------------------------------------------------------------
// WMMA f32 16x16x4 wrapper (CDNA5 V_WMMA_F32_16X16X4_F32)
// ---------------------------------------------------------------------------
__device__ __forceinline__ v8f wmma_f32_16x16x4(v2f a, v2f b, v8f c) {
    // (neg_a, A, neg_b, B, c_mod, C, reuse_a, reuse_b)
    return __builtin_amdgcn_wmma_f32_16x16x4_f32(false, a, false, b, (short)0, c,
                                                 false, false);
}

// ---------------------------------------------------------------------------
// Tiled GEMM: C = epilogue(A[MxK] @ B[KxN])
//   epilogue: optional +bias, optional eval-BatchNorm, optional ReLU
// Block = 128 threads (4 waves), 64x64 output tile, each wave a 32x32 quadrant
// built from 2x2 V_WMMA_F32_16X16X4_F32 accumulators; K staged in LDS (step 16).
// ---------------------------------------------------------------------------
template<bool DO_BN, bool DO_RELU, bool HAS_BIAS>
__global__ __launch_bounds__(128)
void gemm64x64_kernel(const float* __restrict__ A, const float* __restrict__ B,
                      const float* __restrict__ bias,
                      const float* __restrict__ gam, const float* __restrict__ bet,
                      const float* __restrict__ mu,  const float* __restrict__ var,
                      float* __restrict__ C, int M, int K, int N)
{
    __shared__ __align__(16) float As[16 * 65];   // [k][m], padded
    __shared__ __align__(16) float Bs[16 * 68];   // [k][n], padded (float4-friendly)

    const int tid  = threadIdx.x;
    const int lane = tid & 31;
    const int wave = tid >> 5;     // 0..3
    const int wr   = wave >> 1;    // wave row (0..1) -> 32-row strip
    const int wc   = wave & 1;     // wave col (0..1) -> 32-col strip
    const int m0   = blockIdx.y * 64;
    const int n0   = blockIdx.x * 64;
    const int lm   = lane & 15;    // M (A) / N (B) index within 16
    const int lk   = lane >> 4;    // lane-half selects K pair

    v8f acc[2][2];
    #pragma unroll
    for (int i = 0; i < 2; ++i)
        #pragma unroll
        for (int j = 0; j < 2; ++j)
            #pragma unroll
            for (int r = 0; r < 8; ++r) acc[i][j][r] = 0.0f;

    for (int k0 = 0; k0 < K; k0 += 16) {
        // ---- stage A tile 64x16 into As[k][m] (K is always a multiple of 4) ----
        #pragma unroll
        for (int s = 0; s < 2; ++s) {
            int id  = tid + s * 128;        // 0..255
            int row = id >> 2;              // 0..63
            int c4  = (id & 3) << 2;        // 0,4,8,12
            int gm  = m0 + row;
            int gk  = k0 + c4;
            v4f av = {0.0f, 0.0f, 0.0f, 0.0f};
            if (gm < M && gk < K)
                av = *(const v4f*)(A + (size_t)gm * K + gk);
            As[(c4 + 0) * 65 + row] = av.x;
            As[(c4 + 1) * 65 + row] = av.y;
            As[(c4 + 2) * 65 + row] = av.z;
            As[(c4 + 3) * 65 + row] = av.w;
        }
        // ---- stage B tile 16x64 into Bs[k][n] ----
        #pragma unroll
        for (int s = 0; s < 2; ++s) {
            int id = tid + s * 128;
            int kr = id >> 4;               // 0..15
            int nc = (id & 15) << 2;        // 0..60
            int gk = k0 + kr;
            v4f bv = {0.0f, 0.0f, 0.0f, 0.0f};
            if (gk < K)
                bv = *(const v4f*)(B + (size_t)gk * N + n0 + nc);
            *(v4f*)(&Bs[kr * 68 + nc]) = bv;
        }
        __syncthreads();

        #pragma unroll
        for (int kk = 0; kk < 16; kk += 4) {
            const int ks = kk + lk * 2;     // lane-half K base
            v2f af[2], bf[2];
            #pragma unroll
            for (int mt = 0; mt < 2; ++mt) {
                af[mt][0] = As[(ks + 0) * 65 + wr * 32 + mt * 16 + lm];
                af[mt][1] = As[(ks + 1) * 65 + wr * 32 + mt * 16 + lm];
            }
            #pragma unroll
            for (int nt = 0; nt < 2; ++nt) {
                bf[nt][0] = Bs[(ks + 0) * 68 + wc * 32 + nt * 16 + lm];
                bf[nt][1] = Bs[(ks + 1) * 68 + wc * 32 + nt * 16 + lm];
            }
            #pragma unroll
            for (int mt = 0; mt < 2; ++mt)
                #pragma unroll
                for (int nt = 0; nt < 2; ++nt)
                    acc[mt][nt] = wmma_f32_16x16x4(af[mt], bf[nt], acc[mt][nt]);
        }
        __syncthreads();
    }

    // ---- epilogue: y = dot*scale + shift ; optional relu ----
    #pragma unroll
    for (int nt = 0; nt < 2; ++nt) {
        const int col = n0 + wc * 32 + nt * 16 + lm;   // N always multiple of 64 here
        float scale = 1.0f, shift = 0.0f;
        if (HAS_BIAS) shift = bias[col];
        if (DO_BN) {
            float s2 = gam[col] * rsqrtf(var[col] + 1e-5f);
            shift = (shift - mu[col]) * s2 + bet[col];
            scale = s2;
        }
        #pragma unroll
        for (int mt = 0; mt < 2; ++mt) {
            #pragma unroll
            for (int r = 0; r < 8; ++r) {
                int row = m0 + wr * 32 + mt * 16 + lk * 8 + r;  // C/D VGPR layout
                if (row < M) {
                    float x = acc[mt][nt][r] * scale + shift;
                    if (DO_RELU) x = fmaxf(x, 0.0f);
                    C[(size_t)row * N + col] = x;
                }
            }
        }
    }
}

// ---------------------------------------------------------------------------
// Graph helpers
// ---------------------------------------------------------------------------
__global__ void deg_count_kernel(const int* __restrict__ dst, float* __restrict__ deg, int E)
{
    int i = blockIdx.x * blockDim.x + threadIdx.x;
    if (i < E) atomicAdd(&deg[dst[i]], 1.0f);
}

__global__ void dinv_finalize_kernel(float* __restrict__ d, int n)
{
    int i = blockIdx.x * blockDim.x + threadIdx.x;
    if (i < n) d[i] = rsqrtf(d[i] + 1.0f);    // +1 self-loop; deg>0 always
}

// 1 wave = 1 edge; lane moves a float4 of the 128-wide row with f32 atomics.
__global__ void scatter_edges_kernel(const float* __restrict__ xw,
                                     const int* __restrict__ src,
                                     const int* __restrict__ dst,
                                     const float* __restrict__ dinv,
                                     float* __restrict__ acc, int E)
{
    int gt   = blockIdx.x * blockDim.x + threadIdx.x;
    int e    = gt >> 5;
    int lane = gt & 31;
    if (e >= E) return;
    int r = src[e], c = dst[e];
    float nrm = dinv[r] * dinv[c];
    v4f xv = *(const v4f*)(xw + (size_t)r * EMB + lane * 4);
    float* ap = acc + (size_t)c * EMB + lane * 4;
    atomicAdd(ap + 0, xv.x * nrm);
    atomicAdd(ap + 1, xv.y * nrm);
    atomicAdd(ap + 2, xv.z * nrm);
    atomicAdd(ap + 3, xv.w * nrm);
}

// Non-atomic: runs after the scatter for this acc on the same stream.
__global__ void selfloop_kernel(const float* __restrict__ xw,
                                const float* __restrict__ dinv,
                                float* __restrict__ acc, int n)
{
    int i = blockIdx.x * blockDim.x + threadIdx.x;
    if (i < n * EMB) {
        int node = i >> 7;                 // EMB == 128
        float dv = dinv[node];
        acc[i] += xw[i] * dv * dv;
    }
}

// feats = relu(acc + bias_relA + bias_relB)
__global__ void bias_relu_kernel(const float* __restrict__ acc,
                                 const float* __restrict__ bA,
                                 const float* __restrict__ bB,
                                 float* __restrict__ out, int total)
{
    int i = blockIdx.x * blockDim.x + threadIdx.x;
    if (i < total) {
        int c = i & (EMB - 1);
        out[i] = fmaxf(acc[i] + bA[c] + bB[c], 0.0f);
    }
}

// one block per embedding column: mean over 2N rows
__global__ void pool_kernel(const float* __restrict__ fr, const float* __restrict__ fp,
                            float* __restrict__ g, int n)
{
    __shared__ float red[256];
    int c = blockIdx.x;
    float s = 0.0f;
    for (int i = threadIdx.x; i < n; i += blockDim.x)
        s += fr[(size_t)i * EMB + c] + fp[(size_t)i * EMB + c];
    red[threadIdx.x] = s;
    __syncthreads();
    for (int off = 128; off > 0; off >>= 1) {
        if (threadIdx.x < off) red[threadIdx.x] += red[threadIdx.x + off];
        __syncthreads();
    }
    if (threadIdx.x == 0) g[c] = red[0] / (2.0f * (float)n);
}

// out[0] = relu(g @ W1 + b1) @ W2 + b2   (g:[128], W1:[128x256], W2:[256x1])
__global__ void pred_kernel(const float* __restrict__ g, const float* __restrict__ W1,
                            const float* __restrict__ b1, const float* __restrict__ W2,
                            const float* __restrict__ b2, float* __restrict__ out)
{
    __shared__ float red[256];
    int j = threadIdx.x;
    float s = b1[j];
    for (int k = 0; k < EMB; ++k) s += g[k] * W1[k * HIDN + j];
    red[j] = fmaxf(s, 0.0f) * W2[j];
    __syncthreads();
    for (int off = 128; off > 0; off >>= 1) {
        if (j < off) red[j] += red[j + off];
        __syncthreads();
    }
    if (j == 0) out[0] = red[0] + b2[0];
}

// ---------------------------------------------------------------------------
// Orchestration
// ---------------------------------------------------------------------------
extern "C" void kernel_launch(void* const* d_in, const int* in_sizes, int n_in,
                              void* d_out, int out_size, void* d_ws, size_t ws_size,
                              hipStream_t stream)
{
    (void)in_sizes; (void)n_in; (void)out_size; (void)ws_size;
    const int N = NNODES, E = NEDGES;

    const float* x_rna = (const float*)d_in[0];
    const float* x_pro = (const float*)d_in[1];
    const int* edges[4] = { (const int*)d_in[2], (const int*)d_in[3],
                            (const int*)d_in[4], (const int*)d_in[5] };

    // params flattened in setup_inputs() insertion order
    int p = 6;
    const float* enc[2][12];   // W1,b1,g1,be1,m1,v1,W2,b2,g2,be2,m2,v2 per type
    for (int t = 0; t < 2; ++t)
        for (int i = 0; i < 12; ++i) enc[t][i] = (const float*)d_in[p++];
    const float* gW[2][4]; const float* gb[2][4];   // rel order: rr, rp, pr, pp
    for (int l = 0; l < 2; ++l)
        for (int r = 0; r < 4; ++r) {
            gW[l][r] = (const float*)d_in[p++];
            gb[l][r] = (const float*)d_in[p++];
        }
    const float* pW1 = (const float*)d_in[p++];
    const float* pb1 = (const float*)d_in[p++];
    const float* pW2 = (const float*)d_in[p++];
    const float* pb2 = (const float*)d_in[p++];

    // workspace layout (floats): ~128.8 MB total
    float* ws = (float*)d_ws;
    float* feats[2];
    feats[0] = ws;                                   // [N*128] rna embeddings
    feats[1] = feats[0] + (size_t)N * EMB;           // [N*128] protein embeddings
    float* bufH   = feats[1] + (size_t)N * EMB;      // [N*256] encoder hidden / acc pair
    float* accv[2] = { bufH, bufH + (size_t)N * EMB };
    float* xw   = bufH + (size_t)N * HIDN;           // [N*128] per-relation x@W
    float* dinv = xw + (size_t)N * EMB;              // [4*N]   per-relation deg^-1/2
    float* gpool = dinv + (size_t)4 * N;             // [128]

    // 1) per-relation symmetric-normalization terms (targets incl. self-loop)
    for (int r = 0; r < 4; ++r) {
        float* dv = dinv + (size_t)r * N;
        hipMemsetAsync(dv, 0, (size_t)N * sizeof(float), stream);
        deg_count_kernel<<<(E + 255) / 256, 256, 0, stream>>>(edges[r] + E, dv, E);
        dinv_finalize_kernel<<<(N + 255) / 256, 256, 0, stream>>>(dv, N);
    }

    // 2) encoders (GEMM + BN + ReLU, then GEMM + BN)
    dim3 blk(128);
    dim3 g1((HIDN + 63) / 64, (N + 63) / 64);
    dim3 g2((EMB + 63) / 64, (N + 63) / 64);
    gemm64x64_kernel<true, true, true><<<g1, blk, 0, stream>>>(
        x_rna, enc[0][0], enc[0][1], enc[0][2], enc[0][3], enc[0][4], enc[0][5],
        bufH, N, 2000, HIDN);
    gemm64x64_kernel<true, false, true><<<g2, blk, 0, stream>>>(
        bufH, enc[0][6], enc[0][7], enc[0][8], enc[0][9], enc[0][10], enc[0][11],
        feats[0], N, HIDN, EMB);
    gemm64x64_kernel<true, true, true><<<g1, blk, 0, stream>>>(
        x_pro, enc[1][0], enc[1][1], enc[1][2], enc[1][3], enc[1][4], enc[1][5],
        bufH, N, 1000, HIDN);
    gemm64x64_kernel<true, false, true><<<g2, blk, 0, stream>>>(
        bufH, enc[1][6], enc[1][7], enc[1][8], enc[1][9], enc[1][10], enc[1][11],
        feats[1], N, HIDN, EMB);

    // 3) hetero GCN layers
    const int relsrc[4] = {0, 0, 1, 1};    // rr, rp, pr, pp
    const int reldst[4] = {0, 1, 0, 1};
    for (int l = 0; l < 2; ++l) {
        hipMemsetAsync(bufH, 0, (size_t)N * HIDN * sizeof(float), stream);  // both accs
        for (int r = 0; r < 4; ++r) {
            gemm64x64_kernel<false, false, false><<<g2, blk, 0, stream>>>(
                feats[relsrc[r]], gW[l][r], nullptr, nullptr, nullptr, nullptr, nullptr,
                xw, N, EMB, EMB);
            float* dv = dinv + (size_t)r * N;
            scatter_edges_kernel<<<(E * 32 + 255) / 256, 256, 0, stream>>>(
                xw, edges[r], edges[r] + E, dv, accv[reldst[r]], E);
            selfloop_kernel<<<(N * EMB + 255) / 256, 256, 0, stream>>>(
                xw, dv, accv[reldst[r]], N);
        }
        bias_relu_kernel<<<(N * EMB + 255) / 256, 256, 0, stream>>>(
            accv[0], gb[l][0], gb[l][2], feats[0], N * EMB);   // rna <- rr + pr
        bias_relu_kernel<<<(N * EMB + 255) / 256, 256, 0, stream>>>(
            accv[1], gb[l][1], gb[l][3], feats[1], N * EMB);   // protein <- rp + pp
    }

    // 4) mean pool + prediction head
    pool_kernel<<<EMB, 256, 0, stream>>>(feats[0], feats[1], gpool, N);
    pred_kernel<<<1, 256, 0, stream>>>(gpool, pW1, pb1, pW2, pb2, (float*)d_out);
}